// Block_60284160966734
// MI455X (gfx1250) — compile-verified
//
#include <hip/hip_runtime.h>
#include <math.h>

#define BB 4
#define TT 2048
#define CC 32
#define HH 8
#define DD 4
#define EPSV 1e-5f
#define ROWS (BB*TT)      /* 8192 token rows */
#define QT   (TT/16)      /* 128 q-tiles per (b,h) */

typedef __attribute__((ext_vector_type(2))) float v2f;
typedef __attribute__((ext_vector_type(8))) float v8f;

#if defined(__HIP_DEVICE_COMPILE__)
#if !__has_builtin(__builtin_amdgcn_wmma_f32_16x16x4_f32)
#error "missing __builtin_amdgcn_wmma_f32_16x16x4_f32 on this toolchain (device pass)"
#endif
#endif

// Same-wave LDS producer->consumer ordering (K1/K3 row buffers only).
static __device__ __forceinline__ void wave_lds_sync() {
  __builtin_amdgcn_wave_barrier();
  asm volatile("s_wait_dscnt 0" ::: "memory");
  __builtin_amdgcn_wave_barrier();
}

static __device__ __forceinline__ float wsum32(float v) {
#pragma unroll
  for (int off = 16; off > 0; off >>= 1) v += __shfl_xor(v, off, 32);
  return v;
}

// ---------------------------------------------------------------------------
// K1: ln1(x) -> xln ; qkv = xln @ w_attn^T ; scatter to K/Q/V [B,H,T,4]
// ---------------------------------------------------------------------------
__global__ __launch_bounds__(256)
void k1_ln_qkv(const float* __restrict__ x, const float* __restrict__ w_attn,
               const float* __restrict__ g1,
               float* __restrict__ xln, float* __restrict__ kb,
               float* __restrict__ qb, float* __restrict__ vb) {
  __shared__ float wa[3*CC*CC];     // 96 x 32
  __shared__ float xr[8][CC];
  const int tid = threadIdx.x;
  for (int i = tid; i < 3*CC*CC; i += 256) wa[i] = w_attn[i];
  __syncthreads();

  const int w = tid >> 5, lane = tid & 31;
  const int row = blockIdx.x * 8 + w;

  float xv  = x[row*CC + lane];
  float m   = wsum32(xv) * (1.f/CC);
  float d   = xv - m;
  float var = wsum32(d*d) * (1.f/CC);
  float xn  = d * rsqrtf(var + EPSV) * g1[lane];
  xln[row*CC + lane] = xn;
  xr[w][lane] = xn;
  wave_lds_sync();

  float a0 = 0.f, a1 = 0.f, a2 = 0.f;   // split order: k, q, v
#pragma unroll
  for (int k = 0; k < CC; ++k) {
    float xk = xr[w][k];
    a0 = fmaf(xk, wa[(lane        )*CC + k], a0);
    a1 = fmaf(xk, wa[(lane +   CC)*CC + k], a1);
    a2 = fmaf(xk, wa[(lane + 2*CC)*CC + k], a2);
  }
  const int b = row / TT, t = row % TT;
  const int h = lane >> 2, dd = lane & 3;
  const int o = ((b*HH + h)*TT + t)*DD + dd;
  kb[o] = a0; qb[o] = a1; vb[o] = a2;
}

// ---------------------------------------------------------------------------
// K2: causal flash attention in TRANSPOSED score space; one wave per 16-row
// Q tile per (b,h).
//   St = K_tile * (c*Q_tile)^T, c = 0.5*log2(e)  (softmax in exp2 domain)
//   D-layout: reg r -> key j = r+8*half, col -> query i = n.
//   Lane owns query-row i; softmax stats are lane scalars; row reductions =
//   in-lane + one shfl_xor(16).
//   Ot = Vpad^T * Pt  (acc rows = d, cols = i; corr rescale is lane-local).
//   Diagonal tile peeled; K operand software-pipelined one tile ahead.
// ---------------------------------------------------------------------------
__global__ __launch_bounds__(128)
void k2_attn(const float* __restrict__ qb, const float* __restrict__ kb,
             const float* __restrict__ vb, float* __restrict__ att) {
  const int tid = threadIdx.x;
  const int wslot = tid >> 5, lane = tid & 31;
  const int wv = blockIdx.x * 4 + wslot;
  const int it = __builtin_amdgcn_readfirstlane(wv & (QT - 1));  // uniform
  const int bh = __builtin_amdgcn_readfirstlane(wv >> 7);        // uniform
  const int half = lane >> 4, n = lane & 15;
  const int dcl = n & 3;                 // clamped d for branchless V loads
  const float* Q = qb + (size_t)bh*TT*DD;
  const float* K = kb + (size_t)bh*TT*DD;
  const float* V = vb + (size_t)bh*TT*DD;
  const int i0 = it*16;

  // B operand of St: B[k][i] = c*Q[i0+i][k]; lane: i=n, k={2h,2h+1}
  const float qscale = 0.5f * 1.44269504088896340736f;  // 1/sqrt(D) * log2(e)
  v2f bq;
  bq.x = Q[(i0+n)*DD + 2*half] * qscale;
  bq.y = Q[(i0+n)*DD + 2*half + 1] * qscale;

  v8f acc = {0,0,0,0,0,0,0,0};   // Ot: reg r -> row d=r+8h, col i=n
  float mr = -__builtin_inff();  // running max (log2 domain) of row i=n
  float lr = 0.f;                // running sum

  // one 16x16 key tile; scores already in log2 domain
  auto tile = [&](int j0, bool diag, v2f ak) {
    // hoisted V loads (branchless, clamped index; masked later by cndmask)
    float vld[8];
#pragma unroll
    for (int c = 0; c < 4; ++c) {
      const int vr = j0 + 4*c + 2*half;
      vld[2*c]   = V[vr*DD + dcl];
      vld[2*c+1] = V[(vr+1)*DD + dcl];
    }
    v8f z = {0,0,0,0,0,0,0,0};
    v8f st = __builtin_amdgcn_wmma_f32_16x16x4_f32(false, ak, false, bq,
                                                   (short)0, z, false, false);
    float p[8];
    float tmax = -__builtin_inff();
#pragma unroll
    for (int r = 0; r < 8; ++r) {
      const int j = r + 8*half;          // key column held in this reg
      float v = st[r];
      if (diag && j > n) v = -__builtin_inff();
      p[r] = v;
      tmax = fmaxf(tmax, v);
    }
    tmax = fmaxf(tmax, __shfl_xor(tmax, 16, 32));   // combine half-rows
    const float nm   = fmaxf(mr, tmax);
    const float corr = __builtin_amdgcn_exp2f(mr - nm);
    float rs = 0.f;
#pragma unroll
    for (int r = 0; r < 8; ++r) {
      p[r] = __builtin_amdgcn_exp2f(p[r] - nm);
      rs  += p[r];
    }
    rs += __shfl_xor(rs, 16, 32);
    lr = lr*corr + rs;
    mr = nm;
#pragma unroll
    for (int r = 0; r < 8; ++r) acc[r] *= corr;     // lane-local rescale

    // Pt[j][i] lives at reg j%8 of lane i + 16*(j>=8). Pre-swap all regs.
    float sx[8];
#pragma unroll
    for (int r = 0; r < 8; ++r) sx[r] = __shfl_xor(p[r], 16, 32);

    // Ot += Vpad^T(chunk) * Pt(chunk), chunks of 4 over j.
#pragma unroll
    for (int c = 0; c < 4; ++c) {
      v2f av;                            // A[d][k]=V[j0+4c+k][d]; rows>=4 zero
      av.x = (n < DD) ? vld[2*c]   : 0.f;
      av.y = (n < DD) ? vld[2*c+1] : 0.f;
      v2f bp;                            // B[k][i]=Pt[4c+k][i]; j=4c+2h(+1)
      if (c < 2) {   // j in [0,8): source is half-0 registers
        bp.x = half ? sx[4*c + 2] : p[4*c];
        bp.y = half ? sx[4*c + 3] : p[4*c + 1];
      } else {       // j in [8,16): source is half-1 registers
        bp.x = half ? p[4*(c-2) + 2] : sx[4*(c-2)];
        bp.y = half ? p[4*(c-2) + 3] : sx[4*(c-2) + 1];
      }
      acc = __builtin_amdgcn_wmma_f32_16x16x4_f32(false, av, false, bp,
                                                  (short)0, acc, false, false);
    }
  };

  // A operand of St: A[j][k] = K[j0+j][k]; lane: j=n, k={2h,2h+1}
  v2f ak;                                // pipelined one tile ahead
  ak.x = K[n*DD + 2*half];
  ak.y = K[n*DD + 2*half + 1];
  for (int jt = 0; jt < it; ++jt) {      // uniform trip count, no mask
    v2f akn;
    const int j0n = (jt+1)*16;
    akn.x = K[(j0n+n)*DD + 2*half];
    akn.y = K[(j0n+n)*DD + 2*half + 1];
    tile(jt*16, false, ak);
    ak = akn;
  }
  tile(i0, true, ak);                    // peeled diagonal tile (masked)

  // acc regs 0..3 of half 0 hold O[i][0..3]; one float4 store per token.
  if (half == 0) {
    const int b = bh >> 3, h = bh & 7;
    const float inv = 1.f / lr;
    float4 o4 = make_float4(acc[0]*inv, acc[1]*inv, acc[2]*inv, acc[3]*inv);
    float* dst = att + ((size_t)b*TT + (i0 + n))*CC + h*DD;
    *(float4*)dst = o4;
  }
}

// ---------------------------------------------------------------------------
// K3: y=att@w_proj^T+b ; x2=xln+y ; x2n=ln2(x2)*g2 ;
//     h=gelu_exact(x2n@w_ff1^T+b1) ; out=x2n + h@w_ff2^T+b2
// ---------------------------------------------------------------------------
__global__ __launch_bounds__(256)
void k3_proj_ff(const float* __restrict__ xln, const float* __restrict__ attn,
                const float* __restrict__ w_proj, const float* __restrict__ b_proj,
                const float* __restrict__ g2,
                const float* __restrict__ w_ff1, const float* __restrict__ b_ff1,
                const float* __restrict__ w_ff2, const float* __restrict__ b_ff2,
                float* __restrict__ out) {
  __shared__ float wp [CC*CC];      // 32x32
  __shared__ float wf1[4*CC*CC];    // 128x32
  __shared__ float wf2[CC*4*CC];    // 32x128
  __shared__ float arow[8][CC];
  __shared__ float xrow[8][CC];
  __shared__ float hrow[8][4*CC];
  const int tid = threadIdx.x;
  for (int i = tid; i < CC*CC;   i += 256) wp[i]  = w_proj[i];
  for (int i = tid; i < 4*CC*CC; i += 256) wf1[i] = w_ff1[i];
  for (int i = tid; i < 4*CC*CC; i += 256) wf2[i] = w_ff2[i];
  __syncthreads();

  const int w = tid >> 5, lane = tid & 31;
  const int row = blockIdx.x*8 + w;

  arow[w][lane] = attn[row*CC + lane];
  wave_lds_sync();
  float y = b_proj[lane];
#pragma unroll
  for (int k = 0; k < CC; ++k) y = fmaf(arow[w][k], wp[lane*CC + k], y);

  float x2  = xln[row*CC + lane] + y;         // residual from ln1 output
  float m   = wsum32(x2) * (1.f/CC);
  float d   = x2 - m;
  float var = wsum32(d*d) * (1.f/CC);
  float xn  = d * rsqrtf(var + EPSV) * g2[lane];
  xrow[w][lane] = xn;
  wave_lds_sync();

  float h0 = b_ff1[lane], h1 = b_ff1[lane+CC];
  float h2 = b_ff1[lane+2*CC], h3 = b_ff1[lane+3*CC];
#pragma unroll
  for (int k = 0; k < CC; ++k) {
    float xk = xrow[w][k];
    h0 = fmaf(xk, wf1[(lane        )*CC + k], h0);
    h1 = fmaf(xk, wf1[(lane +   CC)*CC + k], h1);
    h2 = fmaf(xk, wf1[(lane + 2*CC)*CC + k], h2);
    h3 = fmaf(xk, wf1[(lane + 3*CC)*CC + k], h3);
  }
  const float is2 = 0.70710678118654752f;     // exact GELU
  h0 = 0.5f*h0*(1.f + erff(h0*is2));
  h1 = 0.5f*h1*(1.f + erff(h1*is2));
  h2 = 0.5f*h2*(1.f + erff(h2*is2));
  h3 = 0.5f*h3*(1.f + erff(h3*is2));
  hrow[w][lane]        = h0;
  hrow[w][lane +   CC] = h1;
  hrow[w][lane + 2*CC] = h2;
  hrow[w][lane + 3*CC] = h3;
  wave_lds_sync();

  float o = b_ff2[lane];
#pragma unroll 4
  for (int k = 0; k < 4*CC; ++k)
    o = fmaf(hrow[w][k], wf2[lane*4*CC + k], o);
  out[row*CC + lane] = xn + o;                // residual from ln2 output
}

// ---------------------------------------------------------------------------
extern "C" void kernel_launch(void* const* d_in, const int* in_sizes, int n_in,
                              void* d_out, int out_size, void* d_ws, size_t ws_size,
                              hipStream_t stream) {
  (void)in_sizes; (void)n_in; (void)out_size; (void)ws_size;
  const float* x      = (const float*)d_in[0];
  const float* w_attn = (const float*)d_in[1];
  const float* w_proj = (const float*)d_in[2];
  const float* b_proj = (const float*)d_in[3];
  const float* g1     = (const float*)d_in[4];
  const float* g2     = (const float*)d_in[5];
  const float* w_ff1  = (const float*)d_in[6];
  const float* b_ff1  = (const float*)d_in[7];
  const float* w_ff2  = (const float*)d_in[8];
  const float* b_ff2  = (const float*)d_in[9];
  float* out = (float*)d_out;
  float* ws  = (float*)d_ws;

  const size_t N = (size_t)ROWS * CC;  // 262144 floats per activation buffer
  float* xln = ws;
  float* kb  = ws + 1*N;
  float* qb  = ws + 2*N;
  float* vb  = ws + 3*N;
  float* att = ws + 4*N;               // total 5 MB workspace

  k1_ln_qkv<<<ROWS/8, 256, 0, stream>>>(x, w_attn, g1, xln, kb, qb, vb);
  k2_attn  <<<(BB*HH*QT)/4, 128, 0, stream>>>(qb, kb, vb, att);
  k3_proj_ff<<<ROWS/8, 256, 0, stream>>>(xln, att, w_proj, b_proj, g2,
                                         w_ff1, b_ff1, w_ff2, b_ff2, out);
}